// Histogram_42760694399478
// MI455X (gfx1250) — compile-verified
//
#include <hip/hip_runtime.h>
#include <stdint.h>

typedef float v2f __attribute__((ext_vector_type(2)));
typedef float v8f __attribute__((ext_vector_type(8)));

#define NBINS        256
#define NCOPIES      64          // LDS-privatized histogram copies (64 KB total)
#define TPB          256         // 8 waves of 32
#define SEGS         8           // workgroups per row
#define PIX_PER_ROW  16384       // 128*128
#define PIX_PER_SEG  (PIX_PER_ROW / SEGS)   // 2048
#define PIX_PER_THR  (PIX_PER_SEG / TPB)    // 8
#define RADIUS       17.0f       // weight < ~1e-9 beyond this |x-k|
#define INV_BW       1.25f       // 1/0.8
#define GROWTH       3.4903429574597902f    // exp(1.25)

__global__ void zero_out_kernel(float* __restrict__ out, int n) {
    int i = blockIdx.x * blockDim.x + threadIdx.x;
    if (i < n) out[i] = 0.0f;
}

__global__ __launch_bounds__(TPB)
void soft_hist_kernel(const float* __restrict__ in, float* __restrict__ out) {
    __shared__ float h[NBINS * NCOPIES];   // h[k*NCOPIES + c], 64 KB

    const int tid  = threadIdx.x;
    const int seg  = blockIdx.x;
    const int row  = blockIdx.y;

    // ---- zero private histograms ----
    #pragma unroll 4
    for (int i = tid; i < NBINS * NCOPIES; i += TPB) h[i] = 0.0f;
    __syncthreads();

    // Flat pointers to LDS carry the group-segment byte offset in the low
    // 32 bits -> raw LDS address for ds_* inline asm.
    const unsigned hbase = (unsigned)(uintptr_t)(&h[0]);

    // ---- main pass: telescoped, windowed sigmoid histogram ----
    const float* src  = in + (size_t)row * PIX_PER_ROW + (size_t)seg * PIX_PER_SEG;
    const int    copy = tid & (NCOPIES - 1);   // bank-conflict-free within a wave

    for (int c4 = 0; c4 < PIX_PER_THR / 4; ++c4) {
        const float4 p = ((const float4*)src)[c4 * TPB + tid];   // coalesced
        float px[4] = {p.x, p.y, p.z, p.w};
        #pragma unroll
        for (int j = 0; j < 4; ++j) {
            const float x    = px[j];
            const int   kmin = (int)fmaxf(ceilf (x - RADIUS), 0.0f);
            const int   kmax = (int)fminf(floorf(x + RADIUS), 255.0f);
            // E_k = sigmoid((x - k + 0.5)/bw) = 1/(1+t_k), t_k = exp((k-0.5-x)/bw)
            float t = __expf(((float)kmin - 0.5f - x) * INV_BW);  // one exp/pixel
            float Eprev = __builtin_amdgcn_rcpf(1.0f + t);
            unsigned a = hbase + (unsigned)(kmin * NCOPIES + copy) * 4u;
            for (int k = kmin; k <= kmax; ++k) {
                t *= GROWTH;                                      // next edge
                const float Enext = __builtin_amdgcn_rcpf(1.0f + t);
                const float w = Eprev - Enext;                    // bin-k weight
                // guaranteed single LDS atomic add (copies shared by 4 waves)
                asm volatile("ds_add_f32 %0, %1" :: "v"(a), "v"(w) : "memory");
                a += NCOPIES * 4u;
                Eprev = Enext;
            }
        }
    }
    // Inline-asm DS ops are not tracked by the compiler's waitcnt insertion:
    // drain this wave's ds_add_f32 before signaling the barrier.
    asm volatile("s_wait_dscnt 0x0" ::: "memory");
    __syncthreads();

    // ---- WMMA reduction over copies: out_part[k] = sum_c h[k][c] ----
    // D = A(16x4, all ones) * B(4 copies x 16 bins) + C, accumulated over c.
    // All-ones A makes the result invariant to the K-slot permutation of B.
    const int wv   = tid >> 5;        // wave id (0..7), each wave reduces 32 bins
    const int lane = tid & 31;
    const int n    = lane & 15;       // N (bin-within-group) index
    const int half = lane >> 4;       // K half select
    v2f a; a.x = 1.0f; a.y = 1.0f;

    #pragma unroll
    for (int g = 0; g < 2; ++g) {
        const int kbase = (wv * 2 + g) * 16;
        v8f acc = {0.f, 0.f, 0.f, 0.f, 0.f, 0.f, 0.f, 0.f};
        #pragma unroll 4
        for (int c = 0; c < NCOPIES; c += 4) {
            const float* p = &h[(kbase + n) * NCOPIES + c + half * 2];
            v2f b; b.x = p[0]; b.y = p[1];
            acc = __builtin_amdgcn_wmma_f32_16x16x4_f32(
                /*neg_a=*/false, a, /*neg_b=*/false, b,
                /*c_mod=*/(short)0, acc, /*reuse_a=*/false, /*reuse_b=*/false);
        }
        // C/D layout: VGPR0, lanes 0..15 -> (M=0, N=lane) = bin sums
        if (lane < 16) {
            __hip_atomic_fetch_add(&out[row * NBINS + kbase + lane], acc[0],
                                   __ATOMIC_RELAXED, __HIP_MEMORY_SCOPE_AGENT);
        }
    }
}

extern "C" void kernel_launch(void* const* d_in, const int* in_sizes, int n_in,
                              void* d_out, int out_size, void* d_ws, size_t ws_size,
                              hipStream_t stream) {
    const float* in  = (const float*)d_in[0];
    float*       out = (float*)d_out;

    // d_out is poisoned by the harness; zero it every call (idempotent launch).
    zero_out_kernel<<<(out_size + 255) / 256, 256, 0, stream>>>(out, out_size);

    const int rows = in_sizes[0] / PIX_PER_ROW;   // 48 for (2,8,3,128,128)
    dim3 grid(SEGS, rows);
    soft_hist_kernel<<<grid, TPB, 0, stream>>>(in, out);
}